// seqMatchNet_36000415875563
// MI455X (gfx1250) — compile-verified
//
#include <hip/hip_runtime.h>
#include <math.h>

typedef __attribute__((ext_vector_type(2))) float v2f;
typedef __attribute__((ext_vector_type(4))) float v4f;
typedef __attribute__((ext_vector_type(8))) float v8f;

#define BATCH 4096
#define SEQL  10
#define DIM   4096
// total rows = BATCH*SEQL = 40960, 16 rows per wave-tile -> 2560 tiles

__global__ void seqmatch_zero_out(float* __restrict__ out) {
  int i = blockIdx.x * blockDim.x + threadIdx.x;
  if (i < BATCH) out[i] = 0.0f;
}

__global__ __launch_bounds__(256)
void seqmatch_diag_wmma(const float* __restrict__ r,
                        const float* __restrict__ c,
                        float* __restrict__ out) {
  const int lane = threadIdx.x & 31;
  const int wave = threadIdx.x >> 5;
  const int tile = blockIdx.x * 8 + wave;          // 16 (b,l) rows per tile
  const long rowBase = (long)tile * 16;

  const int half = lane >> 4;                      // 0: K-slots {0,1}, 1: K-slots {2,3}
  const int m    = lane & 15;                      // row within tile
  const float* __restrict__ rRow = r + (rowBase + m) * (long)DIM + 4 * half;
  const float* __restrict__ cRow = c + (rowBase + m) * (long)DIM + 4 * half;

  v8f acc = {};
  for (int k = 0; k < DIM; k += 8) {
    v4f a = *(const v4f*)(rRow + k);               // global_load_b128
    v4f b = *(const v4f*)(cRow + k);
    // K-slot mapping (order-invariant over K for a dot product):
    //   wmma#1: lanes0-15 slots{0,1}<-cols k,k+1 ; lanes16-31 slots{2,3}<-cols k+4,k+5
    //   wmma#2: lanes0-15 slots{0,1}<-cols k+2,k+3; lanes16-31 slots{2,3}<-cols k+6,k+7
    v2f a0 = {a.x, a.y}, a1 = {a.z, a.w};
    v2f b0 = {b.x, b.y}, b1 = {b.z, b.w};
    acc = __builtin_amdgcn_wmma_f32_16x16x4_f32(false, a0, false, b0,
                                                (short)0, acc, false, false);
    acc = __builtin_amdgcn_wmma_f32_16x16x4_f32(false, a1, false, b1,
                                                (short)0, acc, false, false);
  }

  // Diagonal of 16x16 f32 C/D layout:
  //   lanes 0-7  hold (M=N=lane)      in acc[lane]
  //   lanes 24-31 hold (M=N=lane-16)  in acc[lane-24]
  int sel = -1, rowInTile = 0;
  if (lane < 8)        { sel = lane;      rowInTile = lane;      }
  else if (lane >= 24) { sel = lane - 24; rowInTile = lane - 16; }

  if (sel >= 0) {
    float d = acc[0];
    if (sel == 1) d = acc[1];
    if (sel == 2) d = acc[2];
    if (sel == 3) d = acc[3];
    if (sel == 4) d = acc[4];
    if (sel == 5) d = acc[5];
    if (sel == 6) d = acc[6];
    if (sel == 7) d = acc[7];
    float dist = sqrtf(fmaxf(2.0f - 2.0f * d, 0.0f));
    long grow = rowBase + rowInTile;
    int  b    = (int)(grow / SEQL);
    atomicAdd(out + b, dist * (1.0f / SEQL));      // mean over L
  }
}

extern "C" void kernel_launch(void* const* d_in, const int* in_sizes, int n_in,
                              void* d_out, int out_size, void* d_ws, size_t ws_size,
                              hipStream_t stream) {
  const float* r = (const float*)d_in[0];
  const float* c = (const float*)d_in[1];
  float* out = (float*)d_out;

  seqmatch_zero_out<<<(BATCH + 255) / 256, 256, 0, stream>>>(out);

  const int tiles  = (BATCH * SEQL) / 16;  // 2560
  const int blocks = tiles / 8;            // 320 blocks x 8 waves
  seqmatch_diag_wmma<<<blocks, 256, 0, stream>>>(r, c, out);
}